// SubGraph_6279242187396
// MI455X (gfx1250) — compile-verified
//
#include <hip/hip_runtime.h>
#include <math.h>

typedef __attribute__((ext_vector_type(2))) float v2f;
typedef __attribute__((ext_vector_type(8))) float v8f;

#define NNODES 65536
#define NEDGES 1000000
#define NCLUST 4096
#define POOLW  512
#define HID    64
#define LN_EPS 1e-5f

__device__ __forceinline__ v8f wmma_f32(v2f a, v2f b, v8f c) {
  // D = A(16x4,f32) * B(4x16,f32) + C(16x16,f32)
  return __builtin_amdgcn_wmma_f32_16x16x4_f32(false, a, false, b, (short)0, c,
                                               false, false);
}

// Native CDNA5 float scatter-max atomic (no return -> STOREcnt, drained at
// s_endpgm). IEEE maximumNumber semantics == jax segment_max combine.
__device__ __forceinline__ void atomic_fmax_dev(float* p, float v) {
  asm volatile("global_atomic_max_num_f32 %0, %1, off scope:SCOPE_DEV"
               :: "v"(p), "v"(v)
               : "memory");
}

// ---------------------------------------------------------------------------
// MLP: out[:,0:C) = ReLU(LN(x @ W1 + b1)) @ W2 + b2  ; out rows have stride 2C
// One wave handles 16 nodes. x is [N,C] row-major (stride C).
// ---------------------------------------------------------------------------
__global__ __launch_bounds__(128)
void glp_mlp(const float* __restrict__ x,  const float* __restrict__ W1,
             const float* __restrict__ b1, const float* __restrict__ g,
             const float* __restrict__ beta, const float* __restrict__ W2,
             const float* __restrict__ b2, float* __restrict__ out, int C)
{
  __shared__ float lds[4][16][65];           // padded rows: avoid bank conflicts
  const int lane = threadIdx.x & 31;
  const int wave = threadIdx.x >> 5;
  const int hl   = lane & 15;                // position within half-wave
  const int hi   = lane >> 4;                // 0 = lanes 0-15, 1 = lanes 16-31
  const int base = (blockIdx.x * 4 + wave) * 16;
  const int ldo  = 2 * C;

  // ---------------- GEMM1: [16,C] x [C,64] + b1 ----------------
  v8f acc[4];
#pragma unroll
  for (int nt = 0; nt < 4; ++nt) {
    float bv = b1[nt * 16 + hl];
#pragma unroll
    for (int i = 0; i < 8; ++i) acc[nt][i] = bv;
  }
  // A fragment: lanes 0-15 hold row hl cols (k,k+1); lanes 16-31 cols (k+2,k+3)
  const float* xrow = x + (size_t)(base + hl) * C + 2 * hi;
  for (int k = 0; k < C; k += 4) {
    v2f a = { xrow[k], xrow[k + 1] };
    // B fragment: lanes 0-15 hold (W1[k][col], W1[k+1][col]); hi half k+2,k+3
    const float* wk = W1 + (size_t)(k + 2 * hi) * HID + hl;
#pragma unroll
    for (int nt = 0; nt < 4; ++nt) {
      v2f b = { wk[nt * 16], wk[nt * 16 + HID] };
      acc[nt] = wmma_f32(a, b, acc[nt]);
    }
  }

  // ---------------- LayerNorm + ReLU (rows live across 16 lanes) -------------
  float gv[4], bev[4];
#pragma unroll
  for (int nt = 0; nt < 4; ++nt) { gv[nt] = g[nt*16+hl]; bev[nt] = beta[nt*16+hl]; }
#pragma unroll
  for (int r = 0; r < 8; ++r) {
    float s = acc[0][r] + acc[1][r] + acc[2][r] + acc[3][r];
    float q = acc[0][r]*acc[0][r] + acc[1][r]*acc[1][r]
            + acc[2][r]*acc[2][r] + acc[3][r]*acc[3][r];
#pragma unroll
    for (int m = 1; m <= 8; m <<= 1) {
      s += __shfl_xor(s, m, 32);
      q += __shfl_xor(q, m, 32);
    }
    float mu  = s * (1.0f / 64.0f);
    float var = q * (1.0f / 64.0f) - mu * mu;
    float rs  = rsqrtf(var + LN_EPS);
    int row = r + 8 * hi;
#pragma unroll
    for (int nt = 0; nt < 4; ++nt) {
      float vn = (acc[nt][r] - mu) * rs * gv[nt] + bev[nt];
      lds[wave][row][nt * 16 + hl] = fmaxf(vn, 0.0f);
    }
  }

  // ---------------- GEMM2: [16,64] x [64,C] + b2 ----------------
  const int nt2max = C >> 4;
  for (int nt2 = 0; nt2 < nt2max; ++nt2) {
    const int col = nt2 * 16 + hl;
    float bv = b2[col];
    v8f acc2;
#pragma unroll
    for (int i = 0; i < 8; ++i) acc2[i] = bv;
#pragma unroll
    for (int kk = 0; kk < HID; kk += 4) {
      const float* lp = &lds[wave][hl][kk + 2 * hi];
      v2f a = { lp[0], lp[1] };
      const float* wp = W2 + (size_t)(kk + 2 * hi) * C + col;
      v2f b = { wp[0], wp[C] };
      acc2 = wmma_f32(a, b, acc2);
    }
    float* orow = out + (size_t)(base + 8 * hi) * ldo + col;
#pragma unroll
    for (int r = 0; r < 8; ++r)
      orow[(size_t)r * ldo] = acc2[r];
  }
}

// ---------------------------------------------------------------------------
// Fill agg half of concat buffer with -inf (segment_max identity)
// ---------------------------------------------------------------------------
__global__ void fill_neg_inf(float* __restrict__ xn, int lC, int n)
{
  int t = blockIdx.x * blockDim.x + threadIdx.x;
  if (t >= n) return;
  int C = 1 << lC;
  int row = t >> lC, col = t & (C - 1);
  xn[((size_t)row << (lC + 1)) + C + col] = -__builtin_inff();
}

// ---------------------------------------------------------------------------
// Scatter-max over edges: agg[dst] = max(agg[dst], h[src]) with hw float atomics
// ---------------------------------------------------------------------------
__global__ void edge_scatter(const int* __restrict__ src, const int* __restrict__ dst,
                             float* __restrict__ xn, int lC, int lc4, unsigned total)
{
  unsigned t = blockIdx.x * blockDim.x + threadIdx.x;
  if (t >= total) return;
  unsigned e = t >> lc4;
  int c4 = (int)(t & ((1u << lc4) - 1u)) << 2;
  int C  = 1 << lC;
  int s = src[e], d = dst[e];
  const float4 v = *(const float4*)(xn + ((size_t)s << (lC + 1)) + c4);
  float* ap = xn + ((size_t)d << (lC + 1)) + C + c4;
  atomic_fmax_dev(ap + 0, v.x);
  atomic_fmax_dev(ap + 1, v.y);
  atomic_fmax_dev(ap + 2, v.z);
  atomic_fmax_dev(ap + 3, v.w);
}

// ---------------------------------------------------------------------------
// Replace -inf (no in-edges) with 0 in the agg half
// ---------------------------------------------------------------------------
__global__ void fixup_neg_inf(float* __restrict__ xn, int lC, int n)
{
  int t = blockIdx.x * blockDim.x + threadIdx.x;
  if (t >= n) return;
  int C = 1 << lC;
  int row = t >> lC, col = t & (C - 1);
  float* p = xn + ((size_t)row << (lC + 1)) + C + col;
  float v = *p;
  if (v == -__builtin_inff()) *p = 0.0f;
}

// ---------------------------------------------------------------------------
// Cluster max-pool: clusters are 16 contiguous rows -> pooled [4096,512]
// ---------------------------------------------------------------------------
__global__ void cluster_pool(const float* __restrict__ x3, float* __restrict__ out)
{
  int t = blockIdx.x * blockDim.x + threadIdx.x;
  if (t >= NCLUST * POOLW) return;
  int ci = t >> 9, col = t & (POOLW - 1);
  const float* p = x3 + ((size_t)ci * 16) * POOLW + col;
  float m = -__builtin_inff();
#pragma unroll
  for (int j = 0; j < 16; ++j) m = fmaxf(m, p[(size_t)j * POOLW]);
  out[t] = m;   // clusters are never empty -> no -inf possible
}

// ---------------------------------------------------------------------------
// Per-column L2 norm over 4096 clusters
// ---------------------------------------------------------------------------
__global__ __launch_bounds__(256)
void col_norm(const float* __restrict__ pooled, float* __restrict__ norms)
{
  __shared__ float red[256];
  int col = blockIdx.x;
  float s = 0.0f;
  for (int i = threadIdx.x; i < NCLUST; i += 256) {
    float v = pooled[(size_t)i * POOLW + col];
    s += v * v;
  }
  red[threadIdx.x] = s;
  __syncthreads();
  for (int off = 128; off > 0; off >>= 1) {
    if (threadIdx.x < off) red[threadIdx.x] += red[threadIdx.x + off];
    __syncthreads();
  }
  if (threadIdx.x == 0) norms[col] = sqrtf(red[0]);
}

__global__ void scale_out(float* __restrict__ out, const float* __restrict__ norms)
{
  int t = blockIdx.x * blockDim.x + threadIdx.x;
  if (t >= NCLUST * POOLW) return;
  out[t] = out[t] / (norms[t & (POOLW - 1)] + 1e-6f);
}

// ---------------------------------------------------------------------------
extern "C" void kernel_launch(void* const* d_in, const int* in_sizes, int n_in,
                              void* d_out, int out_size, void* d_ws, size_t ws_size,
                              hipStream_t stream)
{
  const float* x0  = (const float*)d_in[0];
  const int*   ei  = (const int*)d_in[1];
  const int*   src = ei;
  const int*   dst = ei + NEDGES;

  char* ws = (char*)d_ws;
  float* x1    = (float*)(ws);                                         // N*128 f32
  float* x2    = (float*)(ws + (size_t)NNODES * 128 * 4);              // N*256 f32
  float* x3    = (float*)(ws + (size_t)NNODES * (128 + 256) * 4);      // N*512 f32
  float* norms = (float*)(ws + (size_t)NNODES * (128 + 256 + 512) * 4);

  float* outp = (float*)d_out;

  const float* xin = x0;
  float* bufs[3] = { x1, x2, x3 };
  for (int l = 0; l < 3; ++l) {
    int C  = 64 << l;
    int lC = 6 + l;
    const float* W1 = (const float*)d_in[3 + 6 * l + 0];
    const float* b1 = (const float*)d_in[3 + 6 * l + 1];
    const float* g  = (const float*)d_in[3 + 6 * l + 2];
    const float* be = (const float*)d_in[3 + 6 * l + 3];
    const float* W2 = (const float*)d_in[3 + 6 * l + 4];
    const float* b2 = (const float*)d_in[3 + 6 * l + 5];
    float* xn = bufs[l];

    int nAgg = NNODES * C;
    hipLaunchKernelGGL(fill_neg_inf, dim3((nAgg + 255) / 256), dim3(256), 0, stream,
                       xn, lC, nAgg);
    hipLaunchKernelGGL(glp_mlp, dim3(NNODES / 64), dim3(128), 0, stream,
                       xin, W1, b1, g, be, W2, b2, xn, C);
    int lc4 = lC - 2;                               // log2(C/4)
    unsigned total = (unsigned)NEDGES << lc4;       // 16M / 32M / 64M
    hipLaunchKernelGGL(edge_scatter, dim3((total + 255) / 256), dim3(256), 0, stream,
                       src, dst, xn, lC, lc4, total);
    hipLaunchKernelGGL(fixup_neg_inf, dim3((nAgg + 255) / 256), dim3(256), 0, stream,
                       xn, lC, nAgg);
    xin = xn;
  }

  hipLaunchKernelGGL(cluster_pool, dim3((NCLUST * POOLW) / 256), dim3(256), 0, stream,
                     x3, outp);
  hipLaunchKernelGGL(col_norm, dim3(POOLW), dim3(256), 0, stream, outp, norms);
  hipLaunchKernelGGL(scale_out, dim3((NCLUST * POOLW) / 256), dim3(256), 0, stream,
                     outp, norms);
}